// LightGCN_1297080123649
// MI455X (gfx1250) — compile-verified
//
#include <hip/hip_runtime.h>
#include <hip/hip_bf16.h>
#include <math.h>

#define N_NODES 100000
#define N_EDGES 600000
#define IN_FEATS 256
#define H_FEATS 128

typedef __attribute__((ext_vector_type(2))) float v2f;
typedef __attribute__((ext_vector_type(8))) float v8f;

// ---------------------------------------------------------------------------
// Degree accumulation: one thread per edge, native f32 atomics.
// ---------------------------------------------------------------------------
__global__ __launch_bounds__(256) void degree_kernel(
    const int* __restrict__ srcIdx, const int* __restrict__ dstIdx,
    float* __restrict__ degOut, float* __restrict__ degIn)
{
    int e = blockIdx.x * blockDim.x + threadIdx.x;
    if (e < N_EDGES) {
        unsafeAtomicAdd(&degOut[srcIdx[e]], 1.0f);
        unsafeAtomicAdd(&degIn[dstIdx[e]], 1.0f);
    }
}

// In-place: deg -> (deg > 0 ? rsqrt(max(deg,1)) : 0)
__global__ __launch_bounds__(256) void invnorm_kernel(
    float* __restrict__ degOut, float* __restrict__ degIn)
{
    int i = blockIdx.x * blockDim.x + threadIdx.x;
    if (i < N_NODES) {
        float d0 = degOut[i];
        degOut[i] = (d0 > 0.0f) ? rsqrtf(fmaxf(d0, 1.0f)) : 0.0f;
        float d1 = degIn[i];
        degIn[i] = (d1 > 0.0f) ? rsqrtf(fmaxf(d1, 1.0f)) : 0.0f;
    }
}

// ---------------------------------------------------------------------------
// X = A @ W + b  via V_WMMA_F32_16X16X4_F32.
// Block = 256 threads (8 waves). Block computes 32 rows x 128 cols.
// Wave w owns cols [16w, 16w+16) for two 16-row tiles (shares B fragment).
// A fragment (16x4 f32, 2 VGPRs): lanes 0-15 -> K = k,k+1 ; lanes 16-31 -> K = k+2,k+3
// B fragment (4x16 f32, 2 VGPRs): same K split, N = lane%16
// C (16x16 f32, 8 VGPRs): VGPR r, lane l -> M = r + 8*(l/16), N = l%16
// ---------------------------------------------------------------------------
__global__ __launch_bounds__(256) void lin_wmma_kernel(
    const float* __restrict__ A, const float* __restrict__ W,
    const float* __restrict__ bias, float* __restrict__ X)
{
    const int lane   = threadIdx.x & 31;
    const int wave   = threadIdx.x >> 5;   // 0..7
    const int halfId = lane >> 4;          // 0 or 1
    const int l16    = lane & 15;
    const int rowBase = blockIdx.x * 32;
    const int n0      = wave * 16;

    const float* a0p = A + (size_t)(rowBase + l16) * IN_FEATS + 2 * halfId;
    const float* a1p = a0p + (size_t)16 * IN_FEATS;
    const float* bp  = W + (size_t)(2 * halfId) * H_FEATS + n0 + l16;

    v8f c0 = {};
    v8f c1 = {};

    #pragma unroll 8
    for (int k = 0; k < IN_FEATS; k += 4) {
        v2f a0, a1, bf;
        a0.x = a0p[k];
        a0.y = a0p[k + 1];
        a1.x = a1p[k];
        a1.y = a1p[k + 1];
        bf.x = bp[(size_t)k * H_FEATS];
        bf.y = bp[(size_t)(k + 1) * H_FEATS];
        c0 = __builtin_amdgcn_wmma_f32_16x16x4_f32(
                false, a0, false, bf, (short)0, c0, false, false);
        c1 = __builtin_amdgcn_wmma_f32_16x16x4_f32(
                false, a1, false, bf, (short)0, c1, false, false);
    }

    const float bb = bias[n0 + l16];
    #pragma unroll
    for (int r = 0; r < 8; ++r) {
        const int m = r + 8 * halfId;
        X[(size_t)(rowBase + m) * H_FEATS + n0 + l16]      = c0[r] + bb;
        X[(size_t)(rowBase + 16 + m) * H_FEATS + n0 + l16] = c1[r] + bb;
    }
}

// ---------------------------------------------------------------------------
// Edge scatter: one wave per edge; lane holds 4 features (float4 gather,
// 4 native f32 atomic adds into the destination row).
// agg[dst, :] += feat[src, :] * inv_src[src]
// ---------------------------------------------------------------------------
__global__ __launch_bounds__(256) void scatter_kernel(
    const float* __restrict__ feat, const int* __restrict__ srcIdx,
    const int* __restrict__ dstIdx, const float* __restrict__ invSrc,
    float* __restrict__ agg)
{
    const int e = blockIdx.x * 8 + (threadIdx.x >> 5);
    if (e >= N_EDGES) return;
    const int lane = threadIdx.x & 31;
    const int s = srcIdx[e];
    const int d = dstIdx[e];
    const float w = invSrc[s];
    const float4 v = ((const float4*)(feat + (size_t)s * H_FEATS))[lane];
    float* ap = agg + (size_t)d * H_FEATS + lane * 4;
    unsafeAtomicAdd(ap + 0, v.x * w);
    unsafeAtomicAdd(ap + 1, v.y * w);
    unsafeAtomicAdd(ap + 2, v.z * w);
    unsafeAtomicAdd(ap + 3, v.w * w);
}

__device__ __forceinline__ float softmax2(const float* __restrict__ alphas, int idx)
{
    const float a0 = alphas[0], a1 = alphas[1];
    const float m  = fmaxf(a0, a1);
    const float e0 = __expf(a0 - m), e1 = __expf(a1 - m);
    return (idx == 0 ? e0 : e1) / (e0 + e1);
}

// Step 1: h = agg * inv_dst[row]; out = alpha0 * h.  (h overwrites xbuf)
__global__ __launch_bounds__(256) void finalize_first_kernel(
    const float* __restrict__ agg, const float* __restrict__ invDst,
    const float* __restrict__ alphas, float* __restrict__ hbuf,
    float* __restrict__ out)
{
    const int i4 = blockIdx.x * blockDim.x + threadIdx.x;   // float4 index
    const int total4 = (N_NODES * H_FEATS) / 4;
    if (i4 >= total4) return;
    const int n = i4 / (H_FEATS / 4);
    const float s = invDst[n];
    const float a0 = softmax2(alphas, 0);
    float4 v = ((const float4*)agg)[i4];
    v.x *= s; v.y *= s; v.z *= s; v.w *= s;
    ((float4*)hbuf)[i4] = v;
    float4 o;
    o.x = a0 * v.x; o.y = a0 * v.y; o.z = a0 * v.z; o.w = a0 * v.w;
    ((float4*)out)[i4] = o;
}

// Step 2: out += alpha1 * agg * inv_dst[row]
__global__ __launch_bounds__(256) void finalize_second_kernel(
    const float* __restrict__ agg, const float* __restrict__ invDst,
    const float* __restrict__ alphas, float* __restrict__ out)
{
    const int i4 = blockIdx.x * blockDim.x + threadIdx.x;
    const int total4 = (N_NODES * H_FEATS) / 4;
    if (i4 >= total4) return;
    const int n = i4 / (H_FEATS / 4);
    const float s = invDst[n] * softmax2(alphas, 1);
    const float4 v = ((const float4*)agg)[i4];
    float4 o = ((float4*)out)[i4];
    o.x += s * v.x; o.y += s * v.y; o.z += s * v.z; o.w += s * v.w;
    ((float4*)out)[i4] = o;
}

// ---------------------------------------------------------------------------
extern "C" void kernel_launch(void* const* d_in, const int* in_sizes, int n_in,
                              void* d_out, int out_size, void* d_ws, size_t ws_size,
                              hipStream_t stream)
{
    (void)in_sizes; (void)n_in; (void)out_size; (void)ws_size;

    const float* in_feat = (const float*)d_in[0];            // [N_NODES, IN_FEATS]
    const int*   edge    = (const int*)d_in[1];              // [2, N_EDGES]
    const float* W       = (const float*)d_in[2];            // [IN_FEATS, H_FEATS]
    const float* bias    = (const float*)d_in[3];            // [H_FEATS]
    const float* alphas  = (const float*)d_in[4];            // [2]
    float*       out     = (float*)d_out;                    // [N_NODES, H_FEATS]

    const int* srcIdx = edge;
    const int* dstIdx = edge + N_EDGES;

    // Workspace layout (floats): deg_out | deg_in | xbuf(N*H) | agg(N*H)
    float* degOut = (float*)d_ws;
    float* degIn  = degOut + N_NODES;
    float* xbuf   = degIn + N_NODES;
    float* agg    = xbuf + (size_t)N_NODES * H_FEATS;

    const size_t featBytes = (size_t)N_NODES * H_FEATS * sizeof(float);

    // 1) degrees -> normalization factors (in place)
    hipMemsetAsync(degOut, 0, (size_t)2 * N_NODES * sizeof(float), stream);
    degree_kernel<<<(N_EDGES + 255) / 256, 256, 0, stream>>>(srcIdx, dstIdx, degOut, degIn);
    invnorm_kernel<<<(N_NODES + 255) / 256, 256, 0, stream>>>(degOut, degIn);

    // 2) x = in_feat @ W + b   (WMMA fp32)
    lin_wmma_kernel<<<N_NODES / 32, 256, 0, stream>>>(in_feat, W, bias, xbuf);

    // 3) conv step 1: agg = scatter(x * inv_src); h = agg * inv_dst; out = a0*h
    hipMemsetAsync(agg, 0, featBytes, stream);
    scatter_kernel<<<N_EDGES / 8, 256, 0, stream>>>(xbuf, srcIdx, dstIdx, degOut, agg);
    finalize_first_kernel<<<(N_NODES * H_FEATS / 4 + 255) / 256, 256, 0, stream>>>(
        agg, degIn, alphas, xbuf, out);

    // 4) conv step 2: agg = scatter(h * inv_src); out += a1 * agg * inv_dst
    hipMemsetAsync(agg, 0, featBytes, stream);
    scatter_kernel<<<N_EDGES / 8, 256, 0, stream>>>(xbuf, srcIdx, dstIdx, degOut, agg);
    finalize_second_kernel<<<(N_NODES * H_FEATS / 4 + 255) / 256, 256, 0, stream>>>(
        agg, degIn, alphas, out);
}